// GNNLayer_87205015978177
// MI455X (gfx1250) — compile-verified
//
#include <hip/hip_runtime.h>
#include <hip/hip_bf16.h>

#define N_NODES 50000
#define N_EDGES 800000
#define D_IN 128
#define D_OUT 128

typedef float v2f __attribute__((ext_vector_type(2)));
typedef float v8f __attribute__((ext_vector_type(8)));

// ---------------------------------------------------------------- utilities
__global__ void zero_f32_kernel(float* __restrict__ p, int n) {
    int i = blockIdx.x * blockDim.x + threadIdx.x;
    if (i < n) p[i] = 0.0f;
}

// degree[row[e]] += val[e]
__global__ void degree_accum_kernel(const int* __restrict__ row,
                                    const float* __restrict__ val,
                                    float* __restrict__ degree, int n) {
    int e = blockIdx.x * blockDim.x + threadIdx.x;
    if (e < n) atomicAdd(&degree[row[e]], val[e]);
}

// degree -> degree_inv in place (1/d, with 0 -> 0)
__global__ void degree_recip_kernel(float* __restrict__ degree, int n) {
    int i = blockIdx.x * blockDim.x + threadIdx.x;
    if (i < n) {
        float d = degree[i];
        degree[i] = (d != 0.0f) ? (1.0f / d) : 0.0f;
    }
}

// ---------------------------------------------------------------- WMMA GEMM
// support[N_NODES, D_OUT] = x[N_NODES, D_IN] @ w[D_IN, D_OUT], fp32 exact.
// One wave per 16-row tile; 8 x v8f accumulators cover all 128 output cols.
// Fragment layouts per CDNA5 ISA 7.12.2:
//   A 16x4  f32: lane L holds M=L%16, K = 2*(L/16) + vgpr      (float2)
//   B 4x16  f32: vgpr v holds K = v + 2*(L/16), N = L%16       (float2)
//   C 16x16 f32: vgpr r holds M = r + 8*(L/16), N = L%16       (float x8)
__global__ __launch_bounds__(128) void gemm_wmma_kernel(
        const float* __restrict__ x, const float* __restrict__ w,
        float* __restrict__ support) {
    const int lane   = threadIdx.x & 31;
    const int wave   = threadIdx.x >> 5;
    const int tile_m = blockIdx.x * 4 + wave;          // 16-row tile index
    if (tile_m >= (N_NODES / 16)) return;              // wave-uniform exit

    const int m     = tile_m * 16 + (lane & 15);       // A row for this lane
    const int khalf = (lane >> 4) * 2;                 // K sub-offset: 0 or 2
    const int n     = lane & 15;                       // col within 16-tile

    v8f acc[8] = {};                                   // zero accumulators

    const float* __restrict__ arow = x + (size_t)m * D_IN;
    for (int k = 0; k < D_IN; k += 4) {
        v2f a;
        a.x = arow[k + khalf + 0];
        a.y = arow[k + khalf + 1];
        const float* __restrict__ b0 = w + (size_t)(k + khalf + 0) * D_OUT + n;
        const float* __restrict__ b1 = w + (size_t)(k + khalf + 1) * D_OUT + n;
#pragma unroll
        for (int t = 0; t < 8; ++t) {
            v2f b;
            b.x = b0[t * 16];
            b.y = b1[t * 16];
            acc[t] = __builtin_amdgcn_wmma_f32_16x16x4_f32(
                false, a, false, b, (short)0, acc[t], false, false);
        }
    }

    const int crow = tile_m * 16 + (lane >> 4) * 8;    // first row this lane owns
#pragma unroll
    for (int t = 0; t < 8; ++t) {
#pragma unroll
        for (int r = 0; r < 8; ++r) {
            support[(size_t)(crow + r) * D_OUT + t * 16 + n] = acc[t][r];
        }
    }
}

// ---------------------------------------------------------------- scatter
// One wave per edge; lane handles 4 contiguous output features (float4 gather,
// 4 fp32 global atomics). out[row] += support[col] * (val * dinv[row])
__global__ __launch_bounds__(256) void scatter_kernel(
        const int* __restrict__ row, const int* __restrict__ col,
        const float* __restrict__ val, const float* __restrict__ dinv,
        const float* __restrict__ support, float* __restrict__ out) {
    const int lane = threadIdx.x & 31;
    const int e = (blockIdx.x * blockDim.x + threadIdx.x) >> 5;
    if (e >= N_EDGES) return;

    const int r  = row[e];
    const int c  = col[e];
    const float wgt = val[e] * dinv[r];

    const float4 s = *reinterpret_cast<const float4*>(
        support + (size_t)c * D_OUT + lane * 4);
    float* o = out + (size_t)r * D_OUT + lane * 4;
    atomicAdd(o + 0, s.x * wgt);
    atomicAdd(o + 1, s.y * wgt);
    atomicAdd(o + 2, s.z * wgt);
    atomicAdd(o + 3, s.w * wgt);
}

// ---------------------------------------------------------------- finalize
// out = leaky_relu(out + bias, 0.01)
__global__ void finalize_kernel(float* __restrict__ out,
                                const float* __restrict__ bias, int n) {
    int i = blockIdx.x * blockDim.x + threadIdx.x;
    if (i < n) {
        float v = out[i] + bias[i & (D_OUT - 1)];
        out[i] = (v > 0.0f) ? v : 0.01f * v;
    }
}

// ---------------------------------------------------------------- launcher
extern "C" void kernel_launch(void* const* d_in, const int* in_sizes, int n_in,
                              void* d_out, int out_size, void* d_ws, size_t ws_size,
                              hipStream_t stream) {
    const float* x        = (const float*)d_in[0];   // [N_NODES, D_IN]
    const float* weight   = (const float*)d_in[1];   // [D_IN, D_OUT]
    const float* bias     = (const float*)d_in[2];   // [D_OUT]
    const int*   edge_row = (const int*)d_in[3];     // [N_EDGES]
    const int*   edge_col = (const int*)d_in[4];     // [N_EDGES]
    const float* edge_val = (const float*)d_in[5];   // [N_EDGES]
    float* out = (float*)d_out;                      // [N_NODES, D_OUT]

    // workspace: support (25.6 MB) then degree/degree_inv (200 KB)
    float* support = (float*)d_ws;
    float* degree  = support + (size_t)N_NODES * D_OUT;

    const int TB = 256;

    // zero degree and output accumulator
    zero_f32_kernel<<<(N_NODES + TB - 1) / TB, TB, 0, stream>>>(degree, N_NODES);
    zero_f32_kernel<<<(N_NODES * D_OUT + TB - 1) / TB, TB, 0, stream>>>(
        out, N_NODES * D_OUT);

    // degree = segment_sum(edge_val, edge_row); then invert
    degree_accum_kernel<<<(N_EDGES + TB - 1) / TB, TB, 0, stream>>>(
        edge_row, edge_val, degree, N_EDGES);
    degree_recip_kernel<<<(N_NODES + TB - 1) / TB, TB, 0, stream>>>(
        degree, N_NODES);

    // support = x @ W via fp32 WMMA (3125 row-tiles, 4 waves/block)
    const int n_tiles = N_NODES / 16;                // 3125 exact
    gemm_wmma_kernel<<<(n_tiles + 3) / 4, 128, 0, stream>>>(x, weight, support);

    // edge scatter: one wave per edge
    scatter_kernel<<<(N_EDGES * 32) / TB, TB, 0, stream>>>(
        edge_row, edge_col, edge_val, degree, support, out);

    // bias + leaky relu
    finalize_kernel<<<(N_NODES * D_OUT + TB - 1) / TB, TB, 0, stream>>>(
        out, bias, N_NODES * D_OUT);
}